// Node_60241211294274
// MI455X (gfx1250) — compile-verified
//
#include <hip/hip_runtime.h>

typedef __attribute__((ext_vector_type(2))) float v2f;
typedef __attribute__((ext_vector_type(8))) float v8f;

#define N_NODESC 15
#define HID 32
#define KSUB 16
#define FEAT 64
#define WAVES 4
#define BLOCK (WAVES * 32)
#define SPB (WAVES * 16)   // 64 samples per block

// padded LDS strides (floats) chosen for conflict-free 16-lane access
#define XP  68
#define HP  34
#define W1P 18
#define W2P 34

// shared memory layout (float offsets)
#define O_SX   0
#define SZ_SX  (SPB * XP)                 // 4352
#define O_SH   (O_SX + SZ_SX)
#define SZ_SH  (WAVES * 16 * HP)          // 2176
#define O_SW1  (O_SH + SZ_SH)
#define SZ_SW1 (N_NODESC * HID * W1P)     // 8640
#define O_SW2  (O_SW1 + SZ_SW1)
#define SZ_SW2 (N_NODESC * HID * W2P)     // 16320
#define O_SW3  (O_SW2 + SZ_SW2)
#define SZ_SW3 (N_NODESC * 2 * HID)       // 960
#define O_SB1  (O_SW3 + SZ_SW3)
#define O_SB2  (O_SB1 + N_NODESC * HID)
#define O_SB3  (O_SB2 + N_NODESC * HID)
#define O_SIDX (O_SB3 + 32)
#define SMEM_FLOATS (O_SIDX + N_NODESC * KSUB)
#define SMEM_BYTES (SMEM_FLOATS * 4)

__device__ __forceinline__ float leaky1(float v) { return v >= 0.f ? v : 0.01f * v; }

__global__ void __launch_bounds__(BLOCK) tree_mlp_kernel(
    const float* __restrict__ x,  const float* __restrict__ W1, const float* __restrict__ b1,
    const float* __restrict__ W2, const float* __restrict__ b2, const float* __restrict__ W3,
    const float* __restrict__ b3, const float* __restrict__ leaf,
    const int* __restrict__ sidx_g, float* __restrict__ out, int Ntot)
{
    extern __shared__ float smem[];
    float* sX  = smem + O_SX;
    float* sH  = smem + O_SH;
    float* sW1 = smem + O_SW1;
    float* sW2 = smem + O_SW2;
    float* sW3 = smem + O_SW3;
    float* sb1 = smem + O_SB1;
    float* sb2 = smem + O_SB2;
    float* sb3 = smem + O_SB3;
    int*   sidx = (int*)(smem + O_SIDX);

    const int tid = threadIdx.x;

    // ---- stage weights into LDS (padded rows) ----
    for (int i = tid; i < N_NODESC * HID * KSUB; i += BLOCK)
        sW1[(i >> 4) * W1P + (i & 15)] = W1[i];
    for (int i = tid; i < N_NODESC * HID * HID; i += BLOCK)
        sW2[(i >> 5) * W2P + (i & 31)] = W2[i];
    for (int i = tid; i < N_NODESC * 2 * HID; i += BLOCK) sW3[i] = W3[i];
    for (int i = tid; i < N_NODESC * HID; i += BLOCK) { sb1[i] = b1[i]; sb2[i] = b2[i]; }
    for (int i = tid; i < N_NODESC * 2; i += BLOCK) sb3[i] = b3[i];
    for (int i = tid; i < N_NODESC * KSUB; i += BLOCK) sidx[i] = sidx_g[i];

    // ---- stage this block's 64 x-rows (coalesced float4) ----
    const int base = blockIdx.x * SPB;
    for (int q = tid; q < SPB * (FEAT / 4); q += BLOCK) {
        int s = q >> 4, f4 = q & 15;
        int gs = base + s; if (gs > Ntot - 1) gs = Ntot - 1;
        float4 v = ((const float4*)x)[(long)gs * (FEAT / 4) + f4];
        *(float4*)(sX + s * XP + f4 * 4) = v;
    }
    __syncthreads();

    const int wv = tid >> 5;
    const int tl = tid & 31;
    const int n  = tl & 15;        // sample column within tile / A-row low bits
    const int kh = tl >> 4;        // lane half
    const int kk = kh * 2;         // K pair offset for 16x16x4 fragments
    const float* sXn = sX + (wv * 16 + n) * XP;
    float*       sHn = sH + (wv * 16 + n) * HP;

    // zero-mask for layer-3 padded A rows (rows >= 2 of W3pad are zero)
    const float w3mask = (n < 2) ? 1.f : 0.f;

    int loc = 0, off = 0;
    for (int level = 0; level < 4; ++level) {
        const int kc = 1 << level;
        int pending = 0;
        for (int j = 0; j < kc; ++j) {
            const int node = off + j;

            // ---- layer 1: h1(32x16) = W1(32x16) x xs(16x16) ----
            v2f bx[4];
            const int* si = sidx + node * KSUB;
            #pragma unroll
            for (int c = 0; c < 4; ++c) {
                int k = 4 * c + kk;
                bx[c].x = sXn[si[k]];
                bx[c].y = sXn[si[k + 1]];
            }
            v8f h0 = {0.f,0.f,0.f,0.f,0.f,0.f,0.f,0.f};
            v8f h1 = {0.f,0.f,0.f,0.f,0.f,0.f,0.f,0.f};
            {
                const float* w = sW1 + (node * HID + n) * W1P + kk;
                #pragma unroll
                for (int c = 0; c < 4; ++c) {
                    v2f a = *(const v2f*)(w + 4 * c);
                    h0 = __builtin_amdgcn_wmma_f32_16x16x4_f32(false, a, false, bx[c],
                                                               (short)0, h0, false, false);
                }
                w += 16 * W1P;
                #pragma unroll
                for (int c = 0; c < 4; ++c) {
                    v2f a = *(const v2f*)(w + 4 * c);
                    h1 = __builtin_amdgcn_wmma_f32_16x16x4_f32(false, a, false, bx[c],
                                                               (short)0, h1, false, false);
                }
            }
            // bias + leaky, stage h1 column n into LDS (D-layout -> [n][m])
            #pragma unroll
            for (int r = 0; r < 8; ++r) {
                int m0 = r + 8 * kh;
                sHn[m0] = leaky1(h0[r] + sb1[node * HID + m0]);
                int m1 = 16 + m0;
                sHn[m1] = leaky1(h1[r] + sb1[node * HID + m1]);
            }
            asm volatile("" ::: "memory");

            // ---- layer 2: h2(32x16) = W2(32x32) x h1(32x16) ----
            v2f bh[8];
            #pragma unroll
            for (int c = 0; c < 8; ++c)
                bh[c] = *(const v2f*)(sHn + 4 * c + kk);
            asm volatile("" ::: "memory");
            v8f g0 = {0.f,0.f,0.f,0.f,0.f,0.f,0.f,0.f};
            v8f g1 = {0.f,0.f,0.f,0.f,0.f,0.f,0.f,0.f};
            {
                const float* w = sW2 + (node * HID + n) * W2P + kk;
                #pragma unroll
                for (int c = 0; c < 8; ++c) {
                    v2f a = *(const v2f*)(w + 4 * c);
                    g0 = __builtin_amdgcn_wmma_f32_16x16x4_f32(false, a, false, bh[c],
                                                               (short)0, g0, false, false);
                }
                w += 16 * W2P;
                #pragma unroll
                for (int c = 0; c < 8; ++c) {
                    v2f a = *(const v2f*)(w + 4 * c);
                    g1 = __builtin_amdgcn_wmma_f32_16x16x4_f32(false, a, false, bh[c],
                                                               (short)0, g1, false, false);
                }
            }
            #pragma unroll
            for (int r = 0; r < 8; ++r) {
                int m0 = r + 8 * kh;
                sHn[m0] = leaky1(g0[r] + sb2[node * HID + m0]);
                int m1 = 16 + m0;
                sHn[m1] = leaky1(g1[r] + sb2[node * HID + m1]);
            }
            asm volatile("" ::: "memory");

            // ---- layer 3 as WMMA: logits(16x16, rows>=2 zero) = W3pad(16x32) x h2(32x16)
            // A-fragment: unconditional clamped-row loads, then value-level zero mask
            // (keeps control flow straight-line; no EXEC divergence in the hot loop).
            v2f bh2[8];
            #pragma unroll
            for (int c = 0; c < 8; ++c)
                bh2[c] = *(const v2f*)(sHn + 4 * c + kk);
            asm volatile("" ::: "memory");
            v8f lg = {0.f,0.f,0.f,0.f,0.f,0.f,0.f,0.f};
            {
                const float* w3 = sW3 + node * 2 * HID + (n < 2 ? n : 1) * HID + kk;
                #pragma unroll
                for (int c = 0; c < 8; ++c) {
                    v2f t = *(const v2f*)(w3 + 4 * c);  // always safe: row clamped to {0,1}
                    v2f a;
                    a.x = t.x * w3mask;
                    a.y = t.y * w3mask;
                    lg = __builtin_amdgcn_wmma_f32_16x16x4_f32(false, a, false, bh2[c],
                                                               (short)0, lg, false, false);
                }
            }
            // D layout: VGPR r holds row M=r for lanes 0..15 -> lane n has l0=lg[0], l1=lg[1]
            float l0 = lg[0] + sb3[node * 2 + 0];
            float l1 = lg[1] + sb3[node * 2 + 1];
            // p0 < 0.5  <=>  l0 < l1
            int bit = (l0 < l1) ? 1 : 0;
            pending = (loc == j) ? bit : pending;
        }
        loc = 2 * loc + pending;
        off += kc;
    }

    if (tl < 16) {
        long s = (long)base + wv * 16 + n;
        if (s < Ntot) out[s] = leaf[loc];
    }
}

extern "C" void kernel_launch(void* const* d_in, const int* in_sizes, int n_in,
                              void* d_out, int out_size, void* d_ws, size_t ws_size,
                              hipStream_t stream) {
    const float* x    = (const float*)d_in[0];
    const float* W1   = (const float*)d_in[1];
    const float* b1   = (const float*)d_in[2];
    const float* W2   = (const float*)d_in[3];
    const float* b2   = (const float*)d_in[4];
    const float* W3   = (const float*)d_in[5];
    const float* b3   = (const float*)d_in[6];
    const float* leaf = (const float*)d_in[7];
    const int*   sidx = (const int*)d_in[8];
    float* out = (float*)d_out;

    const int Ntot = in_sizes[0] / FEAT;
    (void)n_in; (void)out_size; (void)d_ws; (void)ws_size;

    hipFuncSetAttribute(reinterpret_cast<const void*>(tree_mlp_kernel),
                        hipFuncAttributeMaxDynamicSharedMemorySize, SMEM_BYTES);

    int grid = (Ntot + SPB - 1) / SPB;
    tree_mlp_kernel<<<grid, BLOCK, SMEM_BYTES, stream>>>(
        x, W1, b1, W2, b2, W3, b3, leaf, sidx, out, Ntot);
}